// NeRFRenderer_83846351552922
// MI455X (gfx1250) — compile-verified
//
#include <hip/hip_runtime.h>

#define KBINS 128
#define RAYS_PER_BLOCK 8
#define THREADS (RAYS_PER_BLOCK * 32)

// b128 async copy operates on 16-byte int vectors (per hipcc diagnostic)
typedef int v4i __attribute__((vector_size(16)));
typedef __attribute__((address_space(1))) v4i gv4i_t;   // global
typedef __attribute__((address_space(3))) v4i sv4i_t;   // LDS

__global__ __launch_bounds__(THREADS) void nerf_sample_coarse_kernel(
    const float* __restrict__ weights,
    const float* __restrict__ z_samp,
    const float* __restrict__ uin,
    const float* __restrict__ interp,
    float* __restrict__ out,
    int nRays)
{
    // Per-wave LDS slots (row stride 132 floats = 528 B, 16B-aligned rows)
    __shared__ __align__(16) float wbuf[RAYS_PER_BLOCK][KBINS + 4];
    __shared__ __align__(16) float zbuf[RAYS_PER_BLOCK][KBINS + 4];
    __shared__ __align__(16) float cdfS[RAYS_PER_BLOCK][KBINS + 4];

    const int slot = threadIdx.x >> 5;        // wave index in block
    const int lane = threadIdx.x & 31;        // wave32 lane
    const int ray  = blockIdx.x * RAYS_PER_BLOCK + slot;
    if (ray >= nRays) return;                 // wave-uniform

    const size_t rbase = (size_t)ray * KBINS;
    const float* gw = weights + rbase;
    const float* gz = z_samp  + rbase;

    // ---- Stage weights + z rows into LDS (async DMA on gfx1250) ----
#if __has_builtin(__builtin_amdgcn_global_load_async_to_lds_b128)
    __builtin_amdgcn_global_load_async_to_lds_b128(
        (gv4i_t*)(gw + lane * 4), (sv4i_t*)&wbuf[slot][lane * 4], 0, 0);
    __builtin_amdgcn_global_load_async_to_lds_b128(
        (gv4i_t*)(gz + lane * 4), (sv4i_t*)&zbuf[slot][lane * 4], 0, 0);
#else
    {
        float4 wv = *(const float4*)(gw + lane * 4);
        float4 zv = *(const float4*)(gz + lane * 4);
        *(float4*)&wbuf[slot][lane * 4] = wv;
        *(float4*)&zbuf[slot][lane * 4] = zv;
    }
#endif

    // Overlap: pull u / interval_interp into registers (B128 loads)
    float4 u4 = *(const float4*)(uin    + rbase + lane * 4);
    float4 t4 = *(const float4*)(interp + rbase + lane * 4);

#if __has_builtin(__builtin_amdgcn_global_load_async_to_lds_b128)
#if __has_builtin(__builtin_amdgcn_s_wait_asynccnt)
    __builtin_amdgcn_s_wait_asynccnt(0);
#else
    asm volatile("s_wait_asynccnt 0" ::: "memory");
#endif
#endif
    asm volatile("" ::: "memory");

    // ---- CDF: lane-local partial sums + wave32 inclusive shuffle scan ----
    float4 w4 = *(const float4*)&wbuf[slot][lane * 4];
    float e0 = w4.x + 1e-5f;
    float e1 = w4.y + 1e-5f;
    float e2 = w4.z + 1e-5f;
    float e3 = w4.w + 1e-5f;
    float c0 = e0;
    float c1 = c0 + e1;
    float c2 = c1 + e2;
    float c3 = c2 + e3;

    float incl = c3;
#pragma unroll
    for (int d = 1; d < 32; d <<= 1) {
        float o = __shfl_up(incl, d, 32);
        if (lane >= d) incl += o;
    }
    float total = __shfl(incl, 31, 32);   // sum of all 128 (w + 1e-5)
    float excl  = incl - c3;              // exclusive prefix for this lane
    float inv   = 1.0f / total;

    float4 cv;
    cv.x = (excl + c0) * inv;
    cv.y = (excl + c1) * inv;
    cv.z = (excl + c2) * inv;
    cv.w = (excl + c3) * inv;
    *(float4*)&cdfS[slot][lane * 4] = cv;   // cdfS[k] == cdf[k+1] of reference

    __builtin_amdgcn_wave_barrier();        // per-wave LDS: DS ops are in-order,
    asm volatile("" ::: "memory");          // just stop compiler reordering

    const float* cdf = &cdfS[slot][0];
    const float* zb  = &zbuf[slot][0];

    // ---- Inverse-CDF sample: branchless 7-step binary search + border lerp ----
    float uu[4] = {u4.x, u4.y, u4.z, u4.w};
    float tt[4] = {t4.x, t4.y, t4.z, t4.w};
    float4 res;
    float* rp = &res.x;
#pragma unroll
    for (int j = 0; j < 4; ++j) {
        float uv = uu[j];
        // pos = #entries in cdfS[0..127] that are <= uv  (== searchsorted(cdf,u,right)-1)
        int pos = 0;
#pragma unroll
        for (int step = 64; step >= 1; step >>= 1) {
            pos += (cdf[pos + step - 1] <= uv) ? step : 0;
        }
        int id  = pos > (KBINS - 1) ? (KBINS - 1) : pos;   // clip to [0, K-1]
        int im1 = id > 0 ? id - 1 : 0;
        int ip1 = id < (KBINS - 1) ? id + 1 : (KBINS - 1);
        float zm = zb[im1];
        float zc = zb[id];
        float zp = zb[ip1];
        // border(i) = 0.5*(z[min(i,127)] + z[max(i-1,0)]) reproduces
        // [z0, midpoints, z127] including both edges exactly.
        float left  = 0.5f * (zc + zm);
        float right = 0.5f * (zp + zc);
        float t = tt[j];
        rp[j] = left * (1.0f - t) + right * t;
    }

    *(float4*)(out + rbase + lane * 4) = res;   // coalesced B128 store
}

extern "C" void kernel_launch(void* const* d_in, const int* in_sizes, int n_in,
                              void* d_out, int out_size, void* d_ws, size_t ws_size,
                              hipStream_t stream) {
    // setup_inputs order: rays(unused), weights, z_samp, u, interval_interp
    const float* weights = (const float*)d_in[1];
    const float* z_samp  = (const float*)d_in[2];
    const float* u       = (const float*)d_in[3];
    const float* interp  = (const float*)d_in[4];
    float* out = (float*)d_out;

    const int nRays  = in_sizes[1] / KBINS;
    const int blocks = (nRays + RAYS_PER_BLOCK - 1) / RAYS_PER_BLOCK;
    nerf_sample_coarse_kernel<<<blocks, THREADS, 0, stream>>>(
        weights, z_samp, u, interp, out, nRays);
}